// OneDimensionalCNN_42374147342801
// MI455X (gfx1250) — compile-verified
//
#include <hip/hip_runtime.h>

// ---------------------------------------------------------------------------
// Fused OneDimensionalCNN for MI455X (gfx1250, wave32):
//   conv1d(k=3,SAME) + masked-mean  ==  (prefix-sum row stats) x (GEMM)
//   because both are linear in x. Memory floor = one pass over x (134 MB).
//   GEMMs use V_WMMA_F32_16X16X4_F32 (exact f32, matches reference math).
// ---------------------------------------------------------------------------

typedef float v2f __attribute__((ext_vector_type(2)));
typedef float v8f __attribute__((ext_vector_type(8)));

#define B_    64
#define CIN   512
#define LSEQ  1024
#define COUT  512
#define FOUT  256
#define K1    (3 * CIN)   // 1536: GEMM1 reduction dim (ci,k) flattened
#define K2    COUT        // 512:  GEMM2 reduction dim

// ---------------------------------------------------------------------------
// Kernel 1: per-(b,ci) row reduction. One wave32 per row of x.
//   S  = sum_{l<len} x[row][l]
//   A[b][ci*3+0] = (S - x[len-1])            / len
//   A[b][ci*3+1] =  S                        / len
//   A[b][ci*3+2] = (S - x[0] + x[len]|0)     / len
// ---------------------------------------------------------------------------
__global__ void OneDimCNN_reduce_rows(const float* __restrict__ x,
                                      const int*   __restrict__ list_length,
                                      float*       __restrict__ Amat) {
    const int wave = (blockIdx.x * blockDim.x + threadIdx.x) >> 5; // 0..32767
    const int lane = threadIdx.x & 31;
    const int b    = wave >> 9;                   // wave / CIN
    const float* xp = x + (size_t)wave * LSEQ;

    int len = list_length[b];
    if (len < 1)    len = 1;
    if (len > LSEQ) len = LSEQ;

    // Coalesced float4 loads: 256 float4 per row, 8 per lane.
    const float4* xp4 = (const float4*)xp;
    float s = 0.f;
#pragma unroll
    for (int j = 0; j < LSEQ / 128; ++j) {
        const int i4   = lane + 32 * j;
        const float4 v = xp4[i4];
        const int base = 4 * i4;
        if (base + 0 < len) s += v.x;
        if (base + 1 < len) s += v.y;
        if (base + 2 < len) s += v.z;
        if (base + 3 < len) s += v.w;
    }
    // wave32 butterfly reduction
#pragma unroll
    for (int off = 16; off > 0; off >>= 1)
        s += __shfl_xor(s, off, 32);

    if (lane == 0) {
        const float inv = 1.0f / (float)len;
        const float x0  = xp[0];
        const float xl  = xp[len - 1];
        const float xn  = (len < LSEQ) ? xp[len] : 0.0f;
        float* Ap = Amat + (size_t)wave * 3;   // == b*K1 + ci*3
        Ap[0] = (s - xl)      * inv;   // k = 0
        Ap[1] =  s            * inv;   // k = 1
        Ap[2] = (s - x0 + xn) * inv;   // k = 2
    }
}

// ---------------------------------------------------------------------------
// Kernel 2: H[64,512] = relu( A[64,1536] * conv_w[512,1536]^T + conv_b )
// One wave per 16x16 output tile; V_WMMA_F32_16X16X4_F32 over K=1536.
// A-frag (16x4 f32): lane m (0-15) holds K={0,1}; lane m+16 holds K={2,3}.
// B-frag (4x16 f32): lane n (0-15) holds K={0,1}; lane n+16 holds K={2,3}.
// Both are a float2 load at (row*K + k + 2*(lane>>4)).
// ---------------------------------------------------------------------------
__global__ void OneDimCNN_gemm1_bias_relu(const float* __restrict__ Amat,  // [64, 1536]
                                          const float* __restrict__ W,     // [512, 1536] (conv_w OIW flat)
                                          const float* __restrict__ bias,  // [512]
                                          float*       __restrict__ H) {   // [64, 512]
    const int wave   = (blockIdx.x * blockDim.x + threadIdx.x) >> 5; // 0..127
    const int lane   = threadIdx.x & 31;
    const int tile_n = wave & 31;     // 32 N-tiles (COUT/16)
    const int tile_m = wave >> 5;     // 4  M-tiles (B/16)
    const int half   = lane >> 4;     // K-pair selector
    const int l16    = lane & 15;
    const int m      = tile_m * 16 + l16;
    const int n      = tile_n * 16 + l16;

    const float* Ap = Amat + (size_t)m * K1 + 2 * half;
    const float* Bp = W    + (size_t)n * K1 + 2 * half;

    v8f acc = {};
#pragma unroll 4
    for (int k = 0; k < K1; k += 4) {
        const v2f a = *(const v2f*)(Ap + k);
        const v2f b = *(const v2f*)(Bp + k);
        acc = __builtin_amdgcn_wmma_f32_16x16x4_f32(
            /*neg_a=*/false, a, /*neg_b=*/false, b,
            /*c_mod=*/(short)0, acc, /*reuse_a=*/false, /*reuse_b=*/false);
    }

    // D layout: VGPR r -> M = tile_m*16 + r + 8*half, N = tile_n*16 + l16
    const float bn      = bias[n];
    const int   mg_base = tile_m * 16 + 8 * half;
#pragma unroll
    for (int r = 0; r < 8; ++r) {
        float v = acc[r] + bn;
        v = v > 0.f ? v : 0.f;                    // ReLU
        H[(size_t)(mg_base + r) * COUT + n] = v;
    }
}

// ---------------------------------------------------------------------------
// Kernel 3: out[64,256] = H[64,512] * fc_w[256,512]^T + fc_b
// ---------------------------------------------------------------------------
__global__ void OneDimCNN_gemm2_bias(const float* __restrict__ H,     // [64, 512]
                                     const float* __restrict__ W,     // [256, 512]
                                     const float* __restrict__ bias,  // [256]
                                     float*       __restrict__ out) { // [64, 256]
    const int wave   = (blockIdx.x * blockDim.x + threadIdx.x) >> 5; // 0..63
    const int lane   = threadIdx.x & 31;
    const int tile_n = wave & 15;     // 16 N-tiles (FOUT/16)
    const int tile_m = wave >> 4;     // 4  M-tiles
    const int half   = lane >> 4;
    const int l16    = lane & 15;
    const int m      = tile_m * 16 + l16;
    const int n      = tile_n * 16 + l16;

    const float* Ap = H + (size_t)m * K2 + 2 * half;
    const float* Bp = W + (size_t)n * K2 + 2 * half;

    v8f acc = {};
#pragma unroll 4
    for (int k = 0; k < K2; k += 4) {
        const v2f a = *(const v2f*)(Ap + k);
        const v2f b = *(const v2f*)(Bp + k);
        acc = __builtin_amdgcn_wmma_f32_16x16x4_f32(
            false, a, false, b, (short)0, acc, false, false);
    }

    const float bn      = bias[n];
    const int   mg_base = tile_m * 16 + 8 * half;
#pragma unroll
    for (int r = 0; r < 8; ++r)
        out[(size_t)(mg_base + r) * FOUT + n] = acc[r] + bn;
}

// ---------------------------------------------------------------------------
extern "C" void kernel_launch(void* const* d_in, const int* in_sizes, int n_in,
                              void* d_out, int out_size, void* d_ws, size_t ws_size,
                              hipStream_t stream) {
    const float* x      = (const float*)d_in[0];  // [64, 512, 1024] f32
    const int*   ll     = (const int*)  d_in[1];  // [64] i32
    const float* conv_w = (const float*)d_in[2];  // [512, 512, 3] f32
    const float* conv_b = (const float*)d_in[3];  // [512] f32
    const float* fc_w   = (const float*)d_in[4];  // [256, 512] f32
    const float* fc_b   = (const float*)d_in[5];  // [256] f32
    float*       out    = (float*)d_out;          // [64, 256] f32

    float* Amat = (float*)d_ws;                   // [64, 1536]  (384 KB)
    float* H    = Amat + (size_t)B_ * K1;         // [64, 512]   (128 KB)

    // 1) streaming reduction over x: 32768 rows, 1 wave each, 8 waves/block
    OneDimCNN_reduce_rows<<<(B_ * CIN) / 8, 256, 0, stream>>>(x, ll, Amat);

    // 2) GEMM1 + bias + ReLU: 4x32 = 128 tiles, 8 waves/block
    OneDimCNN_gemm1_bias_relu<<<128 / 8, 256, 0, stream>>>(Amat, conv_w, conv_b, H);

    // 3) GEMM2 + bias: 4x16 = 64 tiles, 8 waves/block
    OneDimCNN_gemm2_bias<<<64 / 8, 256, 0, stream>>>(H, fc_w, fc_b, out);
}